// GraphSAGE_17978733101559
// MI455X (gfx1250) — compile-verified
//
#include <hip/hip_runtime.h>
#include <hip/hip_bf16.h>

typedef float v2f __attribute__((ext_vector_type(2)));
typedef float v8f __attribute__((ext_vector_type(8)));

#define N_SRC 200000
#define N_MID 100000
#define N_TGT 50000
#define D_IN  128
#define D_HID 256
#define D_OUT 64

static __device__ __forceinline__ void atomic_add_f32(float* p, float v) {
    // relaxed, device scope -> global_atomic_add_f32 (no-return variant when result unused)
    __hip_atomic_fetch_add(p, v, __ATOMIC_RELAXED, __HIP_MEMORY_SCOPE_AGENT);
}

// ---------------- scatter layer 1: one wave per edge, 128 floats ----------------
__global__ __launch_bounds__(256) void sage_scatter1(
    const float* __restrict__ x, const int* __restrict__ ei,
    float* __restrict__ agg, float* __restrict__ cnt, int E)
{
    int edge = blockIdx.x * 8 + (threadIdx.x >> 5);
    int lane = threadIdx.x & 31;
    if (edge >= E) return;
    int src = ei[edge];
    int tgt = ei[E + edge];
    float4 v = ((const float4*)(x + (size_t)src * D_IN))[lane];
    float* dst = agg + (size_t)tgt * D_IN + lane * 4;
    atomic_add_f32(dst + 0, v.x);
    atomic_add_f32(dst + 1, v.y);
    atomic_add_f32(dst + 2, v.z);
    atomic_add_f32(dst + 3, v.w);
    if (lane == 0) atomic_add_f32(cnt + tgt, 1.0f);
}

// ---------------- scatter layer 2: one wave per edge, 256 floats ----------------
__global__ __launch_bounds__(256) void sage_scatter2(
    const float* __restrict__ h, const int* __restrict__ ei,
    float* __restrict__ agg, float* __restrict__ cnt, int E)
{
    int edge = blockIdx.x * 8 + (threadIdx.x >> 5);
    int lane = threadIdx.x & 31;
    if (edge >= E) return;
    int src = ei[edge];
    int tgt = ei[E + edge];
    const float4* hs = (const float4*)(h + (size_t)src * D_HID);
    float4 v0 = hs[lane];
    float4 v1 = hs[lane + 32];
    float* d0 = agg + (size_t)tgt * D_HID + lane * 4;
    float* d1 = d0 + 128;
    atomic_add_f32(d0 + 0, v0.x); atomic_add_f32(d0 + 1, v0.y);
    atomic_add_f32(d0 + 2, v0.z); atomic_add_f32(d0 + 3, v0.w);
    atomic_add_f32(d1 + 0, v1.x); atomic_add_f32(d1 + 1, v1.y);
    atomic_add_f32(d1 + 2, v1.z); atomic_add_f32(d1 + 3, v1.w);
    if (lane == 0) atomic_add_f32(cnt + tgt, 1.0f);
}

// ---------------- pack stacked [Wtop; Wbot] (K x N) into B-fragment order ----------------
// frag f = kstep*(N/16)+ntile; lane holds {B[4k+koff][col], B[4k+koff+1][col]},
// koff = lane<16 ? 0 : 2, col = ntile*16 + (lane&15).  (ISA 7.12.2 B layout for 16x16x4 f32)
__global__ __launch_bounds__(256) void sage_packB(
    const float* __restrict__ Wtop, const float* __restrict__ Wbot,
    v2f* __restrict__ out, int K, int N)
{
    int idx = blockIdx.x * blockDim.x + threadIdx.x;
    int total = (K >> 2) * (N >> 4) * 32;
    if (idx >= total) return;
    int lane   = idx & 31;
    int frag   = idx >> 5;
    int ntiles = N >> 4;
    int ntile  = frag % ntiles;
    int kstep  = frag / ntiles;
    int koff   = (lane < 16) ? 0 : 2;
    int row    = kstep * 4 + koff;
    int col    = ntile * 16 + (lane & 15);
    int Khalf  = K >> 1;
    const float* W = (row < Khalf) ? (Wtop + (size_t)row * N)
                                   : (Wbot + (size_t)(row - Khalf) * N);
    v2f r; r.x = W[col]; r.y = W[col + N];
    out[idx] = r;
}

// ---------------- layer-1 GEMM: h = relu([agg/deg | x] @ [Wl1;Wr1] + b1) ----------------
#define LDS1 (D_HID + 4)   // 260: rows 4 banks apart -> conflict-free A-frag b64 reads
__global__ __launch_bounds__(256) void sage_gemm1(
    const float* __restrict__ agg, const float* __restrict__ cnt,
    const float* __restrict__ x, const v2f* __restrict__ Bp,
    const float* __restrict__ bias, float* __restrict__ h)
{
    __shared__ float As[16 * LDS1];
    const int tid = threadIdx.x;
    const int m0 = blockIdx.x * 16;

    // stage A tile: cols 0..127 = agg row / max(cnt,1);  cols 128..255 = x row
    for (int i = tid; i < 16 * 32; i += 256) {
        int row = i >> 5, c4 = i & 31;
        float inv = 1.0f / fmaxf(cnt[m0 + row], 1.0f);
        float4 a = ((const float4*)(agg + (size_t)(m0 + row) * D_IN))[c4];
        float4 u = ((const float4*)(x   + (size_t)(m0 + row) * D_IN))[c4];
        float* da = &As[row * LDS1 + c4 * 4];
        da[0] = a.x * inv; da[1] = a.y * inv; da[2] = a.z * inv; da[3] = a.w * inv;
        float* dx = da + 128;
        dx[0] = u.x; dx[1] = u.y; dx[2] = u.z; dx[3] = u.w;
    }
    __syncthreads();

    const int wave = tid >> 5, lane = tid & 31;
    const int nt0 = wave * 2, nt1 = nt0 + 1;      // 8 waves x 2 N-tiles = 256 cols
    const int arow = lane & 15;
    const int koff = (lane < 16) ? 0 : 2;
    const float* Arow = &As[arow * LDS1 + koff];
    const v2f* B0 = Bp + nt0 * 32 + lane;
    const v2f* B1 = Bp + nt1 * 32 + lane;

    v8f c0 = {}; v8f c1 = {};
    #pragma unroll 4
    for (int ks = 0; ks < 64; ++ks) {             // K = 256, 4 per WMMA
        v2f a  = *(const v2f*)(Arow + ks * 4);
        v2f b0 = B0[ks * 16 * 32];
        v2f b1 = B1[ks * 16 * 32];
        c0 = __builtin_amdgcn_wmma_f32_16x16x4_f32(false, a, false, b0, (short)0, c0, false, false);
        c1 = __builtin_amdgcn_wmma_f32_16x16x4_f32(false, a, false, b1, (short)0, c1, false, false);
    }

    const int rbase = m0 + ((lane < 16) ? 0 : 8);
    const int col0 = nt0 * 16 + (lane & 15);
    const int col1 = col0 + 16;
    const float bb0 = bias[col0], bb1 = bias[col1];
    #pragma unroll
    for (int i = 0; i < 8; ++i) {
        size_t r = (size_t)(rbase + i) * D_HID;
        h[r + col0] = fmaxf(c0[i] + bb0, 0.0f);
        h[r + col1] = fmaxf(c1[i] + bb1, 0.0f);
    }
}

// ---------------- layer-2 GEMM: out = [agg/deg | h] @ [Wl2;Wr2] + b2 ----------------
#define LDS2 (2 * D_HID + 4)   // 516
__global__ __launch_bounds__(128) void sage_gemm2(
    const float* __restrict__ agg, const float* __restrict__ cnt,
    const float* __restrict__ h, const v2f* __restrict__ Bp,
    const float* __restrict__ bias, float* __restrict__ out)
{
    __shared__ float As[16 * LDS2];
    const int tid = threadIdx.x;
    const int m0 = blockIdx.x * 16;

    for (int i = tid; i < 16 * 64; i += 128) {
        int row = i >> 6, c4 = i & 63;
        float inv = 1.0f / fmaxf(cnt[m0 + row], 1.0f);
        float4 a = ((const float4*)(agg + (size_t)(m0 + row) * D_HID))[c4];
        float4 u = ((const float4*)(h   + (size_t)(m0 + row) * D_HID))[c4];
        float* da = &As[row * LDS2 + c4 * 4];
        da[0] = a.x * inv; da[1] = a.y * inv; da[2] = a.z * inv; da[3] = a.w * inv;
        float* dx = da + 256;
        dx[0] = u.x; dx[1] = u.y; dx[2] = u.z; dx[3] = u.w;
    }
    __syncthreads();

    const int wave = tid >> 5, lane = tid & 31;   // wave == N-tile (D_OUT/16 = 4)
    const int arow = lane & 15;
    const int koff = (lane < 16) ? 0 : 2;
    const float* Arow = &As[arow * LDS2 + koff];
    const v2f* B = Bp + wave * 32 + lane;

    v8f c = {};
    #pragma unroll 4
    for (int ks = 0; ks < 128; ++ks) {            // K = 512
        v2f a = *(const v2f*)(Arow + ks * 4);
        v2f b = B[ks * 4 * 32];
        c = __builtin_amdgcn_wmma_f32_16x16x4_f32(false, a, false, b, (short)0, c, false, false);
    }

    const int rbase = m0 + ((lane < 16) ? 0 : 8);
    const int col = wave * 16 + (lane & 15);
    const float bb = bias[col];
    #pragma unroll
    for (int i = 0; i < 8; ++i)
        out[(size_t)(rbase + i) * D_OUT + col] = c[i] + bb;
}

extern "C" void kernel_launch(void* const* d_in, const int* in_sizes, int n_in,
                              void* d_out, int out_size, void* d_ws, size_t ws_size,
                              hipStream_t stream) {
    const float* x   = (const float*)d_in[0];
    const int*   ei1 = (const int*)d_in[1];
    const int*   ei2 = (const int*)d_in[2];
    const float* Wl1 = (const float*)d_in[3];
    const float* Wr1 = (const float*)d_in[4];
    const float* b1  = (const float*)d_in[5];
    const float* Wl2 = (const float*)d_in[6];
    const float* Wr2 = (const float*)d_in[7];
    const float* b2  = (const float*)d_in[8];
    float* out = (float*)d_out;
    const int E1 = in_sizes[1] / 2;
    const int E2 = in_sizes[2] / 2;

    char* ws = (char*)d_ws;
    size_t off = 0;
    auto alloc = [&](size_t bytes) -> void* {
        void* p = ws + off;
        off = (off + bytes + 255) & ~(size_t)255;
        return p;
    };
    float* agg1 = (float*)alloc((size_t)N_MID * D_IN  * 4);
    float* cnt1 = (float*)alloc((size_t)N_MID * 4);
    float* hbuf = (float*)alloc((size_t)N_MID * D_HID * 4);
    float* agg2 = (float*)alloc((size_t)N_TGT * D_HID * 4);
    float* cnt2 = (float*)alloc((size_t)N_TGT * 4);
    v2f*   Bp1  = (v2f*)alloc((size_t)(2 * D_IN)  * D_HID * 4);
    v2f*   Bp2  = (v2f*)alloc((size_t)(2 * D_HID) * D_OUT * 4);

    hipMemsetAsync(agg1, 0, (size_t)N_MID * D_IN  * 4, stream);
    hipMemsetAsync(cnt1, 0, (size_t)N_MID * 4, stream);
    hipMemsetAsync(agg2, 0, (size_t)N_TGT * D_HID * 4, stream);
    hipMemsetAsync(cnt2, 0, (size_t)N_TGT * 4, stream);

    {   // weight repack (tiny, L2-resident)
        int t1 = (2 * D_IN / 4) * (D_HID / 16) * 32;   // 32768
        int t2 = (2 * D_HID / 4) * (D_OUT / 16) * 32;  // 16384
        sage_packB<<<(t1 + 255) / 256, 256, 0, stream>>>(Wl1, Wr1, Bp1, 2 * D_IN,  D_HID);
        sage_packB<<<(t2 + 255) / 256, 256, 0, stream>>>(Wl2, Wr2, Bp2, 2 * D_HID, D_OUT);
    }

    sage_scatter1<<<(E1 + 7) / 8, 256, 0, stream>>>(x, ei1, agg1, cnt1, E1);
    sage_gemm1<<<N_MID / 16, 256, 0, stream>>>(agg1, cnt1, x, Bp1, b1, hbuf);
    sage_scatter2<<<(E2 + 7) / 8, 256, 0, stream>>>(hbuf, ei2, agg2, cnt2, E2);
    sage_gemm2<<<N_TGT / 16, 128, 0, stream>>>(agg2, cnt2, hbuf, Bp2, b2, out);
}